// FastFood_6828998000723
// MI455X (gfx1250) — compile-verified
//
#include <hip/hip_runtime.h>

// CDNA5 / gfx1250 FastFood transform: out = b + S * WHT( G * Perm( WHT( B * x ) ) )
// One workgroup (256 threads = 8 wave32) per batch row; row lives in 32KB LDS.
// WHT_8192 = H16 (x) H16 (x) H16 (x) H2 — the three H16 stages run on
// V_WMMA_F32_16X16X4_F32 (full fp32 precision, K=4 chained x4 per 16x16 tile).

typedef __attribute__((ext_vector_type(2))) float v2f;
typedef __attribute__((ext_vector_type(8))) float v8f;

#define N_DIM 8192
#define THREADS 256

__device__ __forceinline__ v8f h16_wmma(const v2f a[4], const v2f h[4]) {
  v8f d = {};
#pragma unroll
  for (int kk = 0; kk < 4; ++kk) {
    // D = A(16x4) x H(4x16) + D   (SRC0 = data tile, SRC1 = Hadamard signs)
    d = __builtin_amdgcn_wmma_f32_16x16x4_f32(false, a[kk], false, h[kk],
                                              (short)0, d, false, false);
  }
  return d;
}

// In-place unnormalized WHT of ld[0..8191]. Must be called by all 256 threads
// with no divergence (WMMA needs EXEC == all ones). Ends with a barrier.
__device__ __forceinline__ void wht8192(float* ld, int tid, const v2f h[4]) {
  const int lane = tid & 31;
  const int wave = tid >> 5;
  const int m  = lane & 15;        // A-matrix row held by this lane
  const int kh = (lane >> 4) << 1; // K rows held by this half-wave: {kh, kh+1}
  const int mb = (lane >> 4) << 3; // D row offset (0 or 8)
  v2f a[4];
  v8f d;

  // ---- Stage 1: H16 over digit c0 (stride 1). addr = blk*256 + m*16 + k
#pragma unroll
  for (int t4 = 0; t4 < 4; ++t4) {
    const int base = (wave * 4 + t4) * 256;
#pragma unroll
    for (int kk = 0; kk < 4; ++kk) {
      const int k0 = kk * 4 + kh;
      a[kk].x = ld[base + m * 16 + k0];
      a[kk].y = ld[base + m * 16 + k0 + 1];
    }
    d = h16_wmma(a, h);
#pragma unroll
    for (int r = 0; r < 8; ++r)
      ld[base + (r + mb) * 16 + m] = d[r];
  }

  // Stage 2 reads exactly the 256-blocks this wave just wrote:
  // same-wave LDS ops are in-order, so no workgroup barrier needed here.
  // ---- Stage 2: H16 over digit c1 (stride 16). addr = blk*256 + k*16 + m
#pragma unroll
  for (int t4 = 0; t4 < 4; ++t4) {
    const int base = (wave * 4 + t4) * 256;
#pragma unroll
    for (int kk = 0; kk < 4; ++kk) {
      const int k0 = kk * 4 + kh;
      a[kk].x = ld[base + k0 * 16 + m];
      a[kk].y = ld[base + (k0 + 1) * 16 + m];
    }
    d = h16_wmma(a, h);
#pragma unroll
    for (int r = 0; r < 8; ++r)
      ld[base + m * 16 + (r + mb)] = d[r];
  }
  __syncthreads();

  // ---- Stage 3: H16 over digit c2 (stride 256). tile = (c3, c1):
  //      addr = c3*4096 + k*256 + c1*16 + m
#pragma unroll
  for (int t4 = 0; t4 < 4; ++t4) {
    const int t = wave * 4 + t4;
    const int base = (t >> 4) * 4096 + (t & 15) * 16;
#pragma unroll
    for (int kk = 0; kk < 4; ++kk) {
      const int k0 = kk * 4 + kh;
      a[kk].x = ld[base + k0 * 256 + m];
      a[kk].y = ld[base + (k0 + 1) * 256 + m];
    }
    d = h16_wmma(a, h);
#pragma unroll
    for (int r = 0; r < 8; ++r)
      ld[base + m * 256 + (r + mb)] = d[r];
  }
  __syncthreads();

  // ---- Stage 4: H2 over digit c3 (stride 4096) — plain VALU butterflies.
#pragma unroll
  for (int j = 0; j < 16; ++j) {
    const int i = tid + j * THREADS;           // i in [0, 4096)
    const float ua = ld[i];
    const float ub = ld[i + 4096];
    ld[i]        = ua + ub;
    ld[i + 4096] = ua - ub;
  }
  __syncthreads();
}

__global__ __launch_bounds__(THREADS) void fastfood_kernel(
    const float* __restrict__ x, const float* __restrict__ S,
    const float* __restrict__ G, const float* __restrict__ Bsign,
    const float* __restrict__ bvec, const int* __restrict__ P,
    float* __restrict__ out) {
  __shared__ float ld[N_DIM];
  const int tid = threadIdx.x;
  const long long row = blockIdx.x;
  const float* xrow = x + row * N_DIM;
  float* orow = out + row * N_DIM;

  // Per-lane constant Hadamard signs for the WMMA B operand:
  // B[k][n] = (-1)^popc(k & n); lane holds cols n = lane&15, rows {kh, kh+1}.
  const int lane = tid & 31;
  const int hn = lane & 15;
  const int hkh = (lane >> 4) << 1;
  v2f h[4];
#pragma unroll
  for (int kk = 0; kk < 4; ++kk) {
    const int k0 = kk * 4 + hkh;
    h[kk].x = (__popc(k0 & hn) & 1) ? -1.0f : 1.0f;
    h[kk].y = (__popc((k0 + 1) & hn) & 1) ? -1.0f : 1.0f;
  }

  // ---- Phase 1: u = B .* x  (single coalesced read of x, float4)
#pragma unroll
  for (int j = 0; j < 8; ++j) {
    const int i4 = tid + j * THREADS;          // float4 index in [0, 2048)
    const float4 xv = reinterpret_cast<const float4*>(xrow)[i4];
    const float4 bv = reinterpret_cast<const float4*>(Bsign)[i4];
    float4 r;
    r.x = xv.x * bv.x; r.y = xv.y * bv.y;
    r.z = xv.z * bv.z; r.w = xv.w * bv.w;
    reinterpret_cast<float4*>(ld)[i4] = r;
  }
  __syncthreads();

  // ---- Phase 2: v = WHT(u)
  wht8192(ld, tid, h);

  // ---- Phase 3: t[i] = G[i] * v[P[i]]  (gather via registers, then write back)
  float tmp[32];
#pragma unroll
  for (int j = 0; j < 32; ++j) {
    const int i = tid + j * THREADS;
    tmp[j] = G[i] * ld[P[i]];
  }
  __syncthreads();                              // all gathers done before overwrite
#pragma unroll
  for (int j = 0; j < 32; ++j)
    ld[tid + j * THREADS] = tmp[j];
  __syncthreads();

  // ---- Phase 4: z = WHT(t)
  wht8192(ld, tid, h);

  // ---- Phase 5: out = b + S .* z  (single coalesced write, float4)
#pragma unroll
  for (int j = 0; j < 8; ++j) {
    const int i4 = tid + j * THREADS;
    const float4 zv = reinterpret_cast<const float4*>(ld)[i4];
    const float4 sv = reinterpret_cast<const float4*>(S)[i4];
    const float4 bb = reinterpret_cast<const float4*>(bvec)[i4];
    float4 r;
    r.x = bb.x + sv.x * zv.x; r.y = bb.y + sv.y * zv.y;
    r.z = bb.z + sv.z * zv.z; r.w = bb.w + sv.w * zv.w;
    reinterpret_cast<float4*>(orow)[i4] = r;
  }
}

extern "C" void kernel_launch(void* const* d_in, const int* in_sizes, int n_in,
                              void* d_out, int out_size, void* d_ws, size_t ws_size,
                              hipStream_t stream) {
  const float* x     = (const float*)d_in[0];
  const float* S     = (const float*)d_in[1];
  const float* G     = (const float*)d_in[2];
  const float* Bsign = (const float*)d_in[3];
  const float* bvec  = (const float*)d_in[4];
  const int*   P     = (const int*)d_in[5];
  float* out = (float*)d_out;
  const int batch = in_sizes[0] / N_DIM;  // 4096 rows
  hipLaunchKernelGGL(fastfood_kernel, dim3(batch), dim3(THREADS), 0, stream,
                     x, S, G, Bsign, bvec, P, out);
}